// E3Conv_81587198755469
// MI455X (gfx1250) — compile-verified
//
#include <hip/hip_runtime.h>
#include <hip/hip_bf16.h>
#include <math.h>

#define NN 3072
#define NE 98304
#define HH 32
#define NBOND 5

static_assert(sizeof(float) == 4, "");

typedef __attribute__((ext_vector_type(2))) float v2f;
typedef __attribute__((ext_vector_type(4))) float v4f;
typedef __attribute__((ext_vector_type(8))) float v8f;

#define SQ3F 1.7320508075688772f
#define INV_SQ3F 0.5773502691896258f

__device__ __forceinline__ float silu_f(float x) { return x / (1.0f + expf(-x)); }

// ---------------------------------------------------------------------------
// bond init: each thread owns 20 floats (4 bond rows of 5) = 80B aligned block.
// Pattern within 20 floats: positions 0,5,10,15 get 1e6 -> diagonal float4s.
// Non-temporal b128 stores: output is write-only, keep it out of L2.
// ---------------------------------------------------------------------------
__global__ void k_fill_bond(float* __restrict__ out, int nblocks20) {
  int t = blockIdx.x * blockDim.x + threadIdx.x;
  if (t >= nblocks20) return;
  v4f* p = (v4f*)(out + (size_t)t * 20);
  v4f a = {1.0e6f, 0.0f, 0.0f, 0.0f};
  v4f b = {0.0f, 1.0e6f, 0.0f, 0.0f};
  v4f c = {0.0f, 0.0f, 1.0e6f, 0.0f};
  v4f d = {0.0f, 0.0f, 0.0f, 1.0e6f};
  v4f z = {0.0f, 0.0f, 0.0f, 0.0f};
  __builtin_nontemporal_store(a, p + 0);
  __builtin_nontemporal_store(b, p + 1);
  __builtin_nontemporal_store(c, p + 2);
  __builtin_nontemporal_store(d, p + 3);
  __builtin_nontemporal_store(z, p + 4);
}

__global__ void k_zero4(float* __restrict__ p, int n4) {
  int i = blockIdx.x * blockDim.x + threadIdx.x;
  if (i < n4) {
    v4f z = {0.0f, 0.0f, 0.0f, 0.0f};
    ((v4f*)p)[i] = z;
  }
}

// ---------------------------------------------------------------------------
// pack W2/b2 into row-major (K=32) x (Mpad) B matrices for the GEMMs
// Bp[i*576 + m]:  m<512: (j,k2,o) = (m/64, (m/32)&1, m&31), kk = kk_mul*k2+kk_add
//                 m>=512: bias cols
// ---------------------------------------------------------------------------
__global__ void k_pack_B576(const float* __restrict__ w2, const float* __restrict__ b2,
                            float* __restrict__ Bp, int wstride, int kk_mul, int kk_add) {
  int t = blockIdx.x * blockDim.x + threadIdx.x;
  if (t >= 32 * 576) return;
  int i = t / 576;
  int m = t - i * 576;
  float v;
  if (m < 512) {
    int j = m >> 6;
    int k2 = (m >> 5) & 1;
    int o = m & 31;
    int kk = kk_mul * k2 + kk_add;
    v = w2[j * wstride + kk * 1024 + i * 32 + o];
  } else {
    int mm = m - 512;
    int k2 = mm >> 5;
    int o = mm & 31;
    int kk = kk_mul * k2 + kk_add;
    v = b2[kk * 1024 + i * 32 + o];
  }
  Bp[t] = v;
}

// edge-MLP pack: Bp[i*96+m]; m<80: (r,j,o)=(m/40,(m%40)/5,m%5); m in [80,90): bias; else 0
__global__ void k_pack_B96(const float* __restrict__ w2, const float* __restrict__ b2,
                           float* __restrict__ Bp, int ksel) {
  int t = blockIdx.x * blockDim.x + threadIdx.x;
  if (t >= 32 * 96) return;
  int i = t / 96;
  int m = t - i * 96;
  float v = 0.0f;
  if (m < 80) {
    int r = m / 40;
    int mm = m % 40;
    int j = mm / 5;
    int o = mm % 5;
    v = w2[j * 1280 + ksel * 320 + (r * 32 + i) * 5 + o];
  } else if (m < 90) {
    int mm = m - 80;
    int r = mm / 5;
    int o = mm % 5;
    v = b2[ksel * 320 + (r * 32 + i) * 5 + o];
  }
  Bp[t] = v;
}

// ---------------------------------------------------------------------------
// gather node embeddings
// ---------------------------------------------------------------------------
__global__ void k_gather_x(const float* __restrict__ emb, const int* __restrict__ types,
                           float* __restrict__ x) {
  int i = blockIdx.x * blockDim.x + threadIdx.x;
  if (i >= NN * 32) return;
  x[i] = emb[types[i >> 5] * 32 + (i & 31)];
}

// ---------------------------------------------------------------------------
// per-edge precompute: sh(4), soft-one-hot(8), three silu-hidden vectors (8 each)
// ---------------------------------------------------------------------------
__global__ void k_edge_pre(const float* __restrict__ coords, const int* __restrict__ src,
                           const int* __restrict__ dst,
                           const float* __restrict__ w10, const float* __restrict__ b10,
                           const float* __restrict__ w11, const float* __restrict__ b11,
                           const float* __restrict__ w1e, const float* __restrict__ b1e,
                           float* __restrict__ sh4, float* __restrict__ h0,
                           float* __restrict__ h1, float* __restrict__ he) {
  int e = blockIdx.x * blockDim.x + threadIdx.x;
  if (e >= NE) return;
  int sn = src[e], dn = dst[e];
  float vx = coords[dn * 3 + 0] - coords[sn * 3 + 0];
  float vy = coords[dn * 3 + 1] - coords[sn * 3 + 1];
  float vz = coords[dn * 3 + 2] - coords[sn * 3 + 2];
  float r = sqrtf(vx * vx + vy * vy + vz * vz);
  float ir = 1.0f / fmaxf(r, 1e-9f);
  sh4[e * 4 + 0] = 1.0f;
  sh4[e * 4 + 1] = SQ3F * vy * ir;
  sh4[e * 4 + 2] = SQ3F * vz * ir;
  sh4[e * 4 + 3] = SQ3F * vx * ir;

  // soft one-hot: centers (p+1)*step, step = CUTOFF/(EA+1) = 10/9
  const float inv_step = 9.0f / 10.0f;
  float ea[8];
  float dnrm = r * inv_step;
#pragma unroll
  for (int p = 0; p < 8; ++p) {
    float df = dnrm - (float)(p + 1);
    ea[p] = expf(-df * df) * (1.0f / 1.12f);
  }
#pragma unroll
  for (int j = 0; j < 8; ++j) {
    float a0 = b10[j], a1 = b11[j], a2 = b1e[j];
#pragma unroll
    for (int p = 0; p < 8; ++p) {
      a0 = fmaf(ea[p], w10[p * 8 + j], a0);
      a1 = fmaf(ea[p], w11[p * 8 + j], a1);
      a2 = fmaf(ea[p], w1e[p * 8 + j], a2);
    }
    h0[e * 8 + j] = silu_f(a0);
    h1[e * 8 + j] = silu_f(a1);
    he[e * 8 + j] = silu_f(a2);
  }
}

// ---------------------------------------------------------------------------
// fp32 WMMA GEMM:  C[nrows x Mpad] = A[nrows x 32] @ B[32 x Mpad]
// one wave per 16x16 tile, 8 k-steps of V_WMMA_F32_16X16X4_F32
// ---------------------------------------------------------------------------
__global__ __launch_bounds__(256) void k_gemm_k32(const float* __restrict__ A,
                                                  const float* __restrict__ B,
                                                  float* __restrict__ C,
                                                  int nrows, int Mpad) {
  const int lane = threadIdx.x & 31;
  const int wave = threadIdx.x >> 5;
  const int tilesN = Mpad >> 4;
  const int tiles = (nrows >> 4) * tilesN;
  int t = blockIdx.x * 8 + wave;
  if (t >= tiles) return;
  const int tm = t / tilesN;
  const int tn = t - tm * tilesN;
  const int half = lane >> 4;   // lanes 16-31 hold K=2,3 (A) / K rows 2,3 (B)
  const int lr = lane & 15;
  const int arow = (tm * 16 + lr) * 32 + half * 2;
  const int bcol = tn * 16 + lr;
  v8f c = {};
#pragma unroll
  for (int k = 0; k < 8; ++k) {
    v2f a, b;
    a.x = A[arow + k * 4 + 0];
    a.y = A[arow + k * 4 + 1];
    b.x = B[(k * 4 + half * 2 + 0) * Mpad + bcol];
    b.y = B[(k * 4 + half * 2 + 1) * Mpad + bcol];
    c = __builtin_amdgcn_wmma_f32_16x16x4_f32(false, a, false, b, (short)0, c, false, false);
  }
  float* Crow = C + (tm * 16 + half * 8) * Mpad + tn * 16 + lr;
#pragma unroll
  for (int r = 0; r < 8; ++r) Crow[r * Mpad] = c[r];
}

// ---------------------------------------------------------------------------
// conv_initial edge scatter: one wave per edge, lane = output channel o
// ---------------------------------------------------------------------------
__global__ __launch_bounds__(256) void k_conv0(const int* __restrict__ src,
                                               const int* __restrict__ dst,
                                               const float* __restrict__ sh4,
                                               const float* __restrict__ h0,
                                               const float* __restrict__ y0,
                                               float* __restrict__ s_acc,
                                               float* __restrict__ v_acc) {
  int e = blockIdx.x * 8 + (threadIdx.x >> 5);
  if (e >= NE) return;
  int o = threadIdx.x & 31;
  int sn = src[e], dn = dst[e];
  float shs = sh4[e * 4 + 0];
  float sv0 = sh4[e * 4 + 1], sv1 = sh4[e * 4 + 2], sv2 = sh4[e * 4 + 3];
  const float* yb = y0 + sn * 576;
  float t0 = yb[512 + o];  // bias contraction, k=0
  float t1 = yb[544 + o];  // bias contraction, k=1
#pragma unroll
  for (int j = 0; j < 8; ++j) {
    float hj = h0[e * 8 + j];
    t0 = fmaf(hj, yb[j * 64 + o], t0);
    t1 = fmaf(hj, yb[j * 64 + 32 + o], t1);
  }
  const float SC = 0.03125f;  // 1/sqrt(H) * 1/sqrt(AVG_DEG) = 1/32
  atomicAdd(&s_acc[dn * 32 + o], t0 * shs * SC);
  float tv = t1 * SC;
  atomicAdd(&v_acc[0 * NN * 32 + dn * 32 + o], tv * sv0);
  atomicAdd(&v_acc[1 * NN * 32 + dn * 32 + o], tv * sv1);
  atomicAdd(&v_acc[2 * NN * 32 + dn * 32 + o], tv * sv2);
}

// ---------------------------------------------------------------------------
// conv_hidden edge scatter: one wave per edge
// ---------------------------------------------------------------------------
__global__ __launch_bounds__(256) void k_conv1(const int* __restrict__ src,
                                               const int* __restrict__ dst,
                                               const float* __restrict__ sh4,
                                               const float* __restrict__ h1,
                                               const float* __restrict__ ys,
                                               const float* __restrict__ yv,
                                               float* __restrict__ ls_acc,
                                               float* __restrict__ lv_acc) {
  int e = blockIdx.x * 8 + (threadIdx.x >> 5);
  if (e >= NE) return;
  int o = threadIdx.x & 31;
  int sn = src[e], dn = dst[e];
  float shs = sh4[e * 4 + 0];
  float sv[3] = {sh4[e * 4 + 1], sh4[e * 4 + 2], sh4[e * 4 + 3]};
  const float* yr = ys + sn * 576;
  const float* yvp[3] = {yv + 0 * NN * 576 + sn * 576, yv + 1 * NN * 576 + sn * 576,
                         yv + 2 * NN * 576 + sn * 576};
  float ts0 = yr[512 + o];  // w0 . s  (bias)
  float ts2 = yr[544 + o];  // w2 . s  (bias)
  float ta[3], tb[3];
#pragma unroll
  for (int c = 0; c < 3; ++c) {
    ta[c] = yvp[c][512 + o];  // w1 . v_c (bias)
    tb[c] = yvp[c][544 + o];  // w3 . v_c (bias)
  }
#pragma unroll
  for (int j = 0; j < 8; ++j) {
    float hj = h1[e * 8 + j];
    ts0 = fmaf(hj, yr[j * 64 + o], ts0);
    ts2 = fmaf(hj, yr[j * 64 + 32 + o], ts2);
#pragma unroll
    for (int c = 0; c < 3; ++c) {
      ta[c] = fmaf(hj, yvp[c][j * 64 + o], ta[c]);
      tb[c] = fmaf(hj, yvp[c][j * 64 + 32 + o], tb[c]);
    }
  }
  const float SC2 = 0.022097086912079608f;  // 1/sqrt(2H) * 1/sqrt(AVG_DEG)
  float msg_s = SC2 * (ts0 * shs + (ta[0] * sv[0] + ta[1] * sv[1] + ta[2] * sv[2]) * INV_SQ3F);
  atomicAdd(&ls_acc[dn * 32 + o], msg_s);
#pragma unroll
  for (int c = 0; c < 3; ++c) {
    atomicAdd(&lv_acc[c * NN * 32 + dn * 32 + o], SC2 * (ts2 * sv[c] + tb[c] * shs));
  }
}

// ---------------------------------------------------------------------------
// skip connection: s' = s*(1-a_s)+ls*a_s ; v' = v*(1-a_v)+lv*a_v
// ---------------------------------------------------------------------------
__global__ void k_skip(const float* __restrict__ s_acc, const float* __restrict__ ls,
                       const float* __restrict__ v_acc, const float* __restrict__ lv,
                       const float* __restrict__ skip, float* __restrict__ sp,
                       float* __restrict__ vp) {
  int i = blockIdx.x * blockDim.x + threadIdx.x;
  if (i >= NN * 32) return;
  int o = i & 31;
  float as = skip[o];
  float av = skip[32 + o];
  sp[i] = s_acc[i] * (1.0f - as) + ls[i] * as;
#pragma unroll
  for (int c = 0; c < 3; ++c) {
    int idx = c * NN * 32 + i;
    vp[idx] = v_acc[idx] * (1.0f - av) + lv[idx] * av;
  }
}

// ---------------------------------------------------------------------------
// final edge output into bond tensor: one thread per edge (NT scattered stores)
// ---------------------------------------------------------------------------
__global__ void k_edge_out(const int* __restrict__ src, const int* __restrict__ dst,
                           const float* __restrict__ sh4, const float* __restrict__ he,
                           const float* __restrict__ zs, const float* __restrict__ zv,
                           float* __restrict__ out) {
  int e = blockIdx.x * blockDim.x + threadIdx.x;
  if (e >= NE) return;
  int sn = src[e], dn = dst[e];
  float shs = sh4[e * 4 + 0];
  float sv[3] = {sh4[e * 4 + 1], sh4[e * 4 + 2], sh4[e * 4 + 3]};
  float h[8];
#pragma unroll
  for (int j = 0; j < 8; ++j) h[j] = he[e * 8 + j];
  const float* zss = zs + sn * 96;
  const float* zsd = zs + dn * 96;
  const float SC4 = 0.08838834764831845f;  // 1/sqrt(4H)
#pragma unroll
  for (int o = 0; o < 5; ++o) {
    float as_ = zss[80 + o] + zsd[85 + o];  // bias, roles src/dst
#pragma unroll
    for (int j = 0; j < 8; ++j) as_ = fmaf(h[j], zss[j * 5 + o] + zsd[40 + j * 5 + o], as_);
    float avsum = 0.0f;
#pragma unroll
    for (int c = 0; c < 3; ++c) {
      const float* zvs = zv + c * NN * 96 + sn * 96;
      const float* zvd = zv + c * NN * 96 + dn * 96;
      float ac = zvs[80 + o] + zvd[85 + o];
#pragma unroll
      for (int j = 0; j < 8; ++j) ac = fmaf(h[j], zvs[j * 5 + o] + zvd[40 + j * 5 + o], ac);
      avsum = fmaf(ac, sv[c], avsum);
    }
    float val = SC4 * (shs * as_ + avsum * INV_SQ3F);
    __builtin_nontemporal_store(val, &out[(sn * NN + dn) * 5 + o]);
    __builtin_nontemporal_store(val, &out[(dn * NN + sn) * 5 + o]);
  }
}

// ---------------------------------------------------------------------------
// host orchestration
// ---------------------------------------------------------------------------
extern "C" void kernel_launch(void* const* d_in, const int* in_sizes, int n_in,
                              void* d_out, int out_size, void* d_ws, size_t ws_size,
                              hipStream_t stream) {
  (void)in_sizes; (void)n_in; (void)out_size; (void)ws_size;
  const float* coords  = (const float*)d_in[0];
  const float* emb     = (const float*)d_in[1];
  const float* w10     = (const float*)d_in[2];
  const float* b10     = (const float*)d_in[3];
  const float* w20     = (const float*)d_in[4];
  const float* b20     = (const float*)d_in[5];
  const float* w11     = (const float*)d_in[6];
  const float* b11     = (const float*)d_in[7];
  const float* w21     = (const float*)d_in[8];
  const float* b21     = (const float*)d_in[9];
  const float* skip    = (const float*)d_in[10];
  const float* w1e     = (const float*)d_in[11];
  const float* b1e     = (const float*)d_in[12];
  const float* w2e     = (const float*)d_in[13];
  const float* b2e     = (const float*)d_in[14];
  const int*   types   = (const int*)d_in[15];
  const int*   ei      = (const int*)d_in[16];
  const int*   src     = ei;
  const int*   dst     = ei + NE;
  float* out = (float*)d_out;

  // workspace carve-up (floats)
  float* w = (float*)d_ws;
  size_t off = 0;
  float* sh4  = w + off; off += (size_t)NE * 4;
  float* h0   = w + off; off += (size_t)NE * 8;
  float* h1   = w + off; off += (size_t)NE * 8;
  float* he   = w + off; off += (size_t)NE * 8;
  float* x    = w + off; off += (size_t)NN * 32;
  float* y0   = w + off; off += (size_t)NN * 576;
  float* ys   = w + off; off += (size_t)NN * 576;
  float* yv   = w + off; off += (size_t)3 * NN * 576;
  float* zs   = w + off; off += (size_t)NN * 96;
  float* zv   = w + off; off += (size_t)3 * NN * 96;
  float* Bp0  = w + off; off += 32 * 576;
  float* Bp1s = w + off; off += 32 * 576;
  float* Bp1v = w + off; off += 32 * 576;
  float* BpEs = w + off; off += 32 * 96;
  float* BpEv = w + off; off += 32 * 96;
  float* acc  = w + off; off += (size_t)8 * NN * 32;  // s,ls,v[3],lv[3]
  float* s_acc  = acc;
  float* ls_acc = acc + 1 * NN * 32;
  float* v_acc  = acc + 2 * NN * 32;
  float* lv_acc = acc + 5 * NN * 32;
  float* sp   = w + off; off += (size_t)NN * 32;
  float* vp   = w + off; off += (size_t)3 * NN * 32;

  // 1) bond init (vectorized NT b128 stores) + accumulator zero
  const int nblocks20 = (NN * NN * NBOND) / 20;  // exact: 2,359,296
  k_fill_bond<<<(nblocks20 + 255) / 256, 256, 0, stream>>>(out, nblocks20);
  k_zero4<<<((8 * NN * 32 / 4) + 255) / 256, 256, 0, stream>>>(acc, 8 * NN * 32 / 4);

  // 2) pack GEMM B matrices
  k_pack_B576<<<(32 * 576 + 255) / 256, 256, 0, stream>>>(w20, b20, Bp0, 2048, 1, 0);
  k_pack_B576<<<(32 * 576 + 255) / 256, 256, 0, stream>>>(w21, b21, Bp1s, 4096, 2, 0);
  k_pack_B576<<<(32 * 576 + 255) / 256, 256, 0, stream>>>(w21, b21, Bp1v, 4096, 2, 1);
  k_pack_B96<<<(32 * 96 + 255) / 256, 256, 0, stream>>>(w2e, b2e, BpEs, 0);
  k_pack_B96<<<(32 * 96 + 255) / 256, 256, 0, stream>>>(w2e, b2e, BpEv, 1);

  // 3) node embedding gather + per-edge precompute
  k_gather_x<<<(NN * 32 + 255) / 256, 256, 0, stream>>>(emb, types, x);
  k_edge_pre<<<(NE + 255) / 256, 256, 0, stream>>>(coords, src, dst, w10, b10, w11, b11,
                                                   w1e, b1e, sh4, h0, h1, he);

  // 4) conv_initial: y0 = X @ Bp0, then edge scatter
  {
    int tiles = (NN / 16) * (576 / 16);
    k_gemm_k32<<<(tiles + 7) / 8, 256, 0, stream>>>(x, Bp0, y0, NN, 576);
  }
  k_conv0<<<NE / 8, 256, 0, stream>>>(src, dst, sh4, h0, y0, s_acc, v_acc);

  // 5) conv_hidden: ys = s @ Bp1s, yv_c = v_c @ Bp1v, then edge scatter
  {
    int tiles = (NN / 16) * (576 / 16);
    k_gemm_k32<<<(tiles + 7) / 8, 256, 0, stream>>>(s_acc, Bp1s, ys, NN, 576);
    for (int c = 0; c < 3; ++c)
      k_gemm_k32<<<(tiles + 7) / 8, 256, 0, stream>>>(v_acc + (size_t)c * NN * 32, Bp1v,
                                                      yv + (size_t)c * NN * 576, NN, 576);
  }
  k_conv1<<<NE / 8, 256, 0, stream>>>(src, dst, sh4, h1, ys, yv, ls_acc, lv_acc);

  // 6) skip connection
  k_skip<<<(NN * 32 + 255) / 256, 256, 0, stream>>>(s_acc, ls_acc, v_acc, lv_acc, skip, sp, vp);

  // 7) edge MLP node-side GEMMs
  {
    int tiles = (NN / 16) * (96 / 16);
    k_gemm_k32<<<(tiles + 7) / 8, 256, 0, stream>>>(sp, BpEs, zs, NN, 96);
    for (int c = 0; c < 3; ++c)
      k_gemm_k32<<<(tiles + 7) / 8, 256, 0, stream>>>(vp + (size_t)c * NN * 32, BpEv,
                                                      zv + (size_t)c * NN * 96, NN, 96);
  }

  // 8) final edge outputs into bond tensor
  k_edge_out<<<(NE + 255) / 256, 256, 0, stream>>>(src, dst, sh4, he, zs, zv, out);
}